// VoteNetDecoder_16501264351503
// MI455X (gfx1250) — compile-verified
//
#include <hip/hip_runtime.h>
#include <math.h>

// ---------------------------------------------------------------------------
// VoteNet decoder for MI455X (gfx1250, wave32, WMMA).
// GEMMs: V_WMMA_F32_16X16X4_F32, one wave per 16x64 output tile, compile-time
// K (weights padded in workspace so no guards anywhere in the inner loop),
// fused scale/shift(+ReLU) epilogue precomputed from bias/BN params.
// ---------------------------------------------------------------------------

typedef __attribute__((ext_vector_type(2))) float v2f;
typedef __attribute__((ext_vector_type(8))) float v8f;

#define EPSV     1e-5f
#define RADIUS_F 0.3f
#define NSAMPLE  16
#define NPOINT   256
#define BATCH    8
#define NPTS     8192
#define DIM      256

__device__ __forceinline__ v8f wmma_f32_k4(v2f a, v2f b, v8f c) {
  return __builtin_amdgcn_wmma_f32_16x16x4_f32(
      false, a, false, b, (short)0, c, false, false);
}

// ---------------------------------------------------------------------------
// GEMM: Y[b][o][n] = act( scale[o] * (sum_k W[o][k] * X[b][k][n]) + shift[o] )
// K = 4*KSTEPS is compile time; W is (Opad x K) with pad rows zeroed; Nn is a
// multiple of 64. One wave computes a 16x64 tile: A shared across 4 WMMAs.
// EXEC is all-ones throughout (no divergent guards).
// ---------------------------------------------------------------------------
template <int KSTEPS, bool RELU>
__global__ __launch_bounds__(32) void gemm_wmma(
    const float* __restrict__ W,       // Opad x K
    const float* __restrict__ X,       // per batch: K x Nn
    float* __restrict__ Y,             // per batch: (>=Opad rows) x Nn
    int Nn, long xbs, long ybs,
    const float* __restrict__ scale,   // Opad
    const float* __restrict__ shift)   // Opad
{
  constexpr int K = KSTEPS * 4;
  const int lane = threadIdx.x;        // 0..31
  const bool hi  = lane >= 16;
  const int  l   = lane & 15;
  const int  n0  = blockIdx.x * 64;
  const int  o0  = blockIdx.y * 16;

  const float* Xb = X + (long)blockIdx.z * xbs;
  float*       Yb = Y + (long)blockIdx.z * ybs;

  const int kofs = hi ? 2 : 0;                       // lanes 16-31: K+2, K+3
  const float* wp = W + (long)(o0 + l) * K + kofs;   // A rows: M = lane & 15

  v8f acc0 = {0.f,0.f,0.f,0.f,0.f,0.f,0.f,0.f};
  v8f acc1 = acc0, acc2 = acc0, acc3 = acc0;

  #pragma unroll 4
  for (int kk = 0; kk < K; kk += 4) {
    v2f a;
    a.x = wp[kk];
    a.y = wp[kk + 1];
    const float* xr0 = Xb + (long)(kk + kofs) * Nn + n0 + l;
    const float* xr1 = xr0 + Nn;
    __builtin_prefetch(xr0 + 4 * (long)Nn, 0, 1);    // global_prefetch next panel
    v2f b0; b0.x = xr0[0];  b0.y = xr1[0];
    v2f b1; b1.x = xr0[16]; b1.y = xr1[16];
    v2f b2; b2.x = xr0[32]; b2.y = xr1[32];
    v2f b3; b3.x = xr0[48]; b3.y = xr1[48];
    acc0 = wmma_f32_k4(a, b0, acc0);
    acc1 = wmma_f32_k4(a, b1, acc1);
    acc2 = wmma_f32_k4(a, b2, acc2);
    acc3 = wmma_f32_k4(a, b3, acc3);
  }

  #pragma unroll
  for (int r = 0; r < 8; ++r) {
    const int row = o0 + r + (hi ? 8 : 0);           // C/D: M = r (+8 hi half)
    const float sc = scale[row];
    const float sh = shift[row];
    float* yr = Yb + (long)row * Nn + n0 + l;        // C/D: N = lane & 15
    float v0 = fmaf(acc0[r], sc, sh);
    float v1 = fmaf(acc1[r], sc, sh);
    float v2 = fmaf(acc2[r], sc, sh);
    float v3 = fmaf(acc3[r], sc, sh);
    if (RELU) {
      v0 = fmaxf(v0, 0.f); v1 = fmaxf(v1, 0.f);
      v2 = fmaxf(v2, 0.f); v3 = fmaxf(v3, 0.f);
    }
    yr[0] = v0; yr[16] = v1; yr[32] = v2; yr[48] = v3;
  }
}

// ---------------------------------------------------------------------------
// Fold bias + BN into per-channel scale/shift:
//   y = (acc + bias - mean) * gam*rsqrt(var+eps) + bet = acc*scale + shift
// Pad rows (o >= O) get scale=shift=0 so padded GEMM rows store 0.
// ---------------------------------------------------------------------------
__global__ void prep_scaleshift(
    const float* __restrict__ bias, const float* __restrict__ gam,
    const float* __restrict__ bet, const float* __restrict__ mean,
    const float* __restrict__ var,
    float* __restrict__ scale, float* __restrict__ shift, int O, int Opad)
{
  const int o = blockIdx.x * 256 + threadIdx.x;
  if (o >= Opad) return;
  if (o >= O) { scale[o] = 0.f; shift[o] = 0.f; return; }
  const float sc = gam ? gam[o] * rsqrtf(var[o] + EPSV) : 1.f;
  const float b  = bias ? bias[o] : 0.f;
  const float sh = (b - (mean ? mean[o] : 0.f)) * sc + (bet ? bet[o] : 0.f);
  scale[o] = sc; shift[o] = sh;
}

// Pad a (Osrc x Ksrc) weight into (Odst x Kdst), zero-filling extras.
__global__ void pad_weight(const float* __restrict__ src, float* __restrict__ dst,
                           int Osrc, int Ksrc, int Odst, int Kdst)
{
  const long id = (long)blockIdx.x * 256 + threadIdx.x;
  if (id >= (long)Odst * Kdst) return;
  const int o = (int)(id / Kdst), k = (int)(id % Kdst);
  dst[id] = (o < Osrc && k < Ksrc) ? src[(long)o * Ksrc + k] : 0.f;
}

// ---------------------------------------------------------------------------
// vote epilogue: vote_xyz = seed_xyz + net[0:3];  vf = normalize(seed_feat^T
// + net[3:259]).  net is (B, 272(alloc), N), rows 0..258 valid.
// ---------------------------------------------------------------------------
__global__ __launch_bounds__(256) void vote_epilogue(
    const float* __restrict__ net,
    const float* __restrict__ seed_xyz,
    const float* __restrict__ seed_feat,
    float* __restrict__ vote_xyz,   // (B, N, 3)
    float* __restrict__ vf)         // (B, N, 256)
{
  __shared__ float red[256];
  const int n = blockIdx.x, b = blockIdx.y, c = threadIdx.x;
  const float* nb = net + (long)b * 272 * NPTS;

  const float val = seed_feat[((long)b * DIM + c) * NPTS + n]
                  + nb[(long)(3 + c) * NPTS + n];
  red[c] = val * val;
  __syncthreads();
  for (int s = 128; s > 0; s >>= 1) {
    if (c < s) red[c] += red[c + s];
    __syncthreads();
  }
  const float nrm = sqrtf(red[0]);
  vf[((long)b * NPTS + n) * DIM + c] = val / nrm;

  if (c < 3)
    vote_xyz[((long)b * NPTS + n) * 3 + c] =
        seed_xyz[((long)b * NPTS + n) * 3 + c] + nb[(long)c * NPTS + n];
}

// ---------------------------------------------------------------------------
// Farthest point sampling: one block (8 waves) per batch; 256 serial steps;
// dist[] in LDS; argmax ties -> lowest index (matches jnp.argmax).
// ---------------------------------------------------------------------------
__global__ __launch_bounds__(256) void fps_kernel(
    const float* __restrict__ vxyz,      // (B, N, 3)
    float* __restrict__ new_xyz)         // (B, NPOINT, 3) -> d_out
{
  __shared__ float dist[NPTS];
  __shared__ float sd[8];
  __shared__ int   si[8];
  __shared__ int   s_far;

  const int b = blockIdx.x, tid = threadIdx.x;
  const float* xb = vxyz + (long)b * NPTS * 3;

  for (int i = tid; i < NPTS; i += 256) dist[i] = 1e10f;
  int far = 0;
  __syncthreads();

  for (int t = 0; t < NPOINT; ++t) {
    if (tid < 3)
      new_xyz[((long)b * NPOINT + t) * 3 + tid] = xb[(long)far * 3 + tid];

    const float cx = xb[far * 3 + 0];
    const float cy = xb[far * 3 + 1];
    const float cz = xb[far * 3 + 2];

    float best = -1.f; int bidx = 0;
    for (int i = tid; i < NPTS; i += 256) {
      const float dx = xb[i * 3 + 0] - cx;
      const float dy = xb[i * 3 + 1] - cy;
      const float dz = xb[i * 3 + 2] - cz;
      const float nd = fminf(dist[i], dx * dx + dy * dy + dz * dz);
      dist[i] = nd;
      if (nd > best) { best = nd; bidx = i; }   // strict > keeps earliest i
    }
    for (int off = 16; off > 0; off >>= 1) {    // wave32 argmax reduce
      const float ob = __shfl_down(best, off, 32);
      const int   oi = __shfl_down(bidx, off, 32);
      if (ob > best || (ob == best && oi < bidx)) { best = ob; bidx = oi; }
    }
    if ((tid & 31) == 0) { sd[tid >> 5] = best; si[tid >> 5] = bidx; }
    __syncthreads();
    if (tid == 0) {
      float bb = sd[0]; int bi = si[0];
      for (int w = 1; w < 8; ++w)
        if (sd[w] > bb || (sd[w] == bb && si[w] < bi)) { bb = sd[w]; bi = si[w]; }
      s_far = bi;
    }
    __syncthreads();
    far = s_far;
  }
}

// ---------------------------------------------------------------------------
// Ball query: one wave per (b, p); first NSAMPLE lowest-index points with
// d2 < R^2 via ballot prefix-compaction; pad with first hit.
// ---------------------------------------------------------------------------
__global__ __launch_bounds__(32) void ball_query_kernel(
    const float* __restrict__ vxyz,
    const float* __restrict__ new_xyz,
    int* __restrict__ ball_idx)          // (B, NPOINT, NSAMPLE)
{
  __shared__ int slist[NSAMPLE];
  const int p = blockIdx.x, b = blockIdx.y, lane = threadIdx.x;
  const float* xb = vxyz + (long)b * NPTS * 3;
  const float* q  = new_xyz + ((long)b * NPOINT + p) * 3;
  const float qx = q[0], qy = q[1], qz = q[2];
  const float r2 = RADIUS_F * RADIUS_F;

  int total = 0;
  for (int base = 0; base < NPTS && total < NSAMPLE; base += 32) {
    const int n = base + lane;
    const float dx = xb[n * 3 + 0] - qx;
    const float dy = xb[n * 3 + 1] - qy;
    const float dz = xb[n * 3 + 2] - qz;
    const bool ok = (dx * dx + dy * dy + dz * dz) < r2;
    const unsigned long long m = __ballot(ok);
    const int pos = __popcll(m & ((1ull << lane) - 1ull));
    if (ok && (total + pos) < NSAMPLE) slist[total + pos] = n;
    total += __popcll(m);
  }
  __syncthreads();
  if (lane < NSAMPLE) {
    const int t = total < NSAMPLE ? total : NSAMPLE;  // t >= 1 (self in radius)
    ball_idx[((long)b * NPOINT + p) * NSAMPLE + lane] =
        (lane < t) ? slist[lane] : slist[0];
  }
}

// ---------------------------------------------------------------------------
// Gather groups into g (B, 260(pad), NPOINT*NSAMPLE):
//   ch 0..2: (vote_xyz[idx]-new_xyz)/R; ch 3..258: vf[idx]; ch 259: zeros.
// ---------------------------------------------------------------------------
__global__ __launch_bounds__(256) void gather_group_kernel(
    const float* __restrict__ vxyz, const float* __restrict__ vf,
    const float* __restrict__ new_xyz, const int* __restrict__ ball_idx,
    float* __restrict__ g)
{
  __shared__ int sidx[NSAMPLE];
  const int p = blockIdx.x, b = blockIdx.y, c = threadIdx.x;
  if (c < NSAMPLE) sidx[c] = ball_idx[((long)b * NPOINT + p) * NSAMPLE + c];
  __syncthreads();

  const long M  = (long)NPOINT * NSAMPLE;
  const long gb = (long)b * 260 * M;
  for (int s = 0; s < NSAMPLE; ++s) {
    const int idx = sidx[s];
    g[gb + (long)(3 + c) * M + (long)p * NSAMPLE + s] =
        vf[((long)b * NPTS + idx) * DIM + c];
    if (c < 3) {
      const float qv = new_xyz[((long)b * NPOINT + p) * 3 + c];
      g[gb + (long)c * M + (long)p * NSAMPLE + s] =
          (vxyz[((long)b * NPTS + idx) * 3 + c] - qv) * (1.0f / RADIUS_F);
    }
    if (c == 0)  // zero the K-pad channel so padded GEMM K is exact
      g[gb + 259l * M + (long)p * NSAMPLE + s] = 0.f;
  }
}

// ---------------------------------------------------------------------------
// features[b][o][p] = max over s of h2[b][o][p*NSAMPLE + s]
// ---------------------------------------------------------------------------
__global__ __launch_bounds__(256) void maxpool_kernel(
    const float* __restrict__ h2, float* __restrict__ out)
{
  const long id = (long)blockIdx.x * 256 + threadIdx.x;
  if (id >= (long)BATCH * DIM * NPOINT) return;
  const long bo = id / NPOINT;
  const int  p  = (int)(id % NPOINT);
  const float* src = h2 + bo * ((long)NPOINT * NSAMPLE) + (long)p * NSAMPLE;
  float m = src[0];
  #pragma unroll
  for (int s = 1; s < NSAMPLE; ++s) m = fmaxf(m, src[s]);
  out[id] = m;
}

// ---------------------------------------------------------------------------
extern "C" void kernel_launch(void* const* d_in, const int* in_sizes, int n_in,
                              void* d_out, int out_size, void* d_ws, size_t ws_size,
                              hipStream_t stream) {
  const float* seed_xyz  = (const float*)d_in[0];
  const float* seed_feat = (const float*)d_in[1];
  const float* w1 = (const float*)d_in[2];
  const float* b1 = (const float*)d_in[3];
  const float* g1 = (const float*)d_in[4];
  const float* be1 = (const float*)d_in[5];
  const float* m1 = (const float*)d_in[6];
  const float* v1 = (const float*)d_in[7];
  const float* w2 = (const float*)d_in[8];
  const float* b2 = (const float*)d_in[9];
  const float* g2 = (const float*)d_in[10];
  const float* be2 = (const float*)d_in[11];
  const float* m2 = (const float*)d_in[12];
  const float* v2 = (const float*)d_in[13];
  const float* w3 = (const float*)d_in[14];
  const float* b3 = (const float*)d_in[15];
  const float* mw0 = (const float*)d_in[16];
  const float* mg0 = (const float*)d_in[17];
  const float* mb0 = (const float*)d_in[18];
  const float* mm0 = (const float*)d_in[19];
  const float* mv0 = (const float*)d_in[20];
  const float* mw1 = (const float*)d_in[21];
  const float* mg1 = (const float*)d_in[22];
  const float* mb1 = (const float*)d_in[23];
  const float* mm1 = (const float*)d_in[24];
  const float* mv1 = (const float*)d_in[25];
  const float* mw2 = (const float*)d_in[26];
  const float* mg2 = (const float*)d_in[27];
  const float* mb2 = (const float*)d_in[28];
  const float* mm2 = (const float*)d_in[29];
  const float* mv2 = (const float*)d_in[30];

  const int M2 = NPOINT * NSAMPLE;   // 4096

  // workspace carve-up (floats)
  float* bufA = (float*)d_ws;                           // (B,272,N): y1, later net
  float* bufB = bufA + (size_t)BATCH * 272 * NPTS;      // (B,256,N): y2
  float* vf   = bufB + (size_t)BATCH * 256 * NPTS;      // (B,N,256)
  float* vxyz = vf   + (size_t)BATCH * NPTS * DIM;      // (B,N,3)
  int*   bidx = (int*)(vxyz + (size_t)BATCH * NPTS * 3);// (B,NPOINT,NSAMPLE)
  float* gbuf = (float*)(bidx + (size_t)BATCH * NPOINT * NSAMPLE); // (B,260,4096)
  float* h0   = gbuf + (size_t)BATCH * 260 * M2;        // (B,128,4096)
  float* h1   = h0   + (size_t)BATCH * 128 * M2;        // (B,128,4096)
  float* h2   = h1   + (size_t)BATCH * 128 * M2;        // (B,256,4096)
  float* w3p  = h2   + (size_t)BATCH * 256 * M2;        // 272 x 256
  float* mw0p = w3p  + (size_t)272 * 256;               // 128 x 260
  float* ss   = mw0p + (size_t)128 * 260;               // scale/shift pool
  float *sc1 = ss,        *sh1 = sc1 + 256;
  float *sc2 = sh1 + 256, *sh2 = sc2 + 256;
  float *sc3 = sh2 + 256, *sh3 = sc3 + 272;
  float *scm0 = sh3 + 272, *shm0 = scm0 + 128;
  float *scm1 = shm0 + 128, *shm1 = scm1 + 128;
  float *scm2 = shm1 + 128, *shm2 = scm2 + 256;

  float* out_newxyz = (float*)d_out;                        // (B,NPOINT,3)
  float* out_feat   = out_newxyz + (size_t)BATCH * NPOINT * 3;

  // --- prep: pad weights, fold bias/BN into scale/shift -------------------
  pad_weight<<<dim3((272 * 256 + 255) / 256), dim3(256), 0, stream>>>(
      w3, w3p, 259, 256, 272, 256);
  pad_weight<<<dim3((128 * 260 + 255) / 256), dim3(256), 0, stream>>>(
      mw0, mw0p, 128, 259, 128, 260);
  prep_scaleshift<<<dim3(1), dim3(256), 0, stream>>>(b1, g1, be1, m1, v1, sc1, sh1, 256, 256);
  prep_scaleshift<<<dim3(1), dim3(256), 0, stream>>>(b2, g2, be2, m2, v2, sc2, sh2, 256, 256);
  prep_scaleshift<<<dim3(2), dim3(256), 0, stream>>>(b3, nullptr, nullptr, nullptr, nullptr, sc3, sh3, 259, 272);
  prep_scaleshift<<<dim3(1), dim3(256), 0, stream>>>(nullptr, mg0, mb0, mm0, mv0, scm0, shm0, 128, 128);
  prep_scaleshift<<<dim3(1), dim3(256), 0, stream>>>(nullptr, mg1, mb1, mm1, mv1, scm1, shm1, 128, 128);
  prep_scaleshift<<<dim3(1), dim3(256), 0, stream>>>(nullptr, mg2, mb2, mm2, mv2, scm2, shm2, 256, 256);

  const dim3 wblk(32);

  // --- stage 1: three WMMA GEMMs over 8192 points -------------------------
  gemm_wmma<64, true><<<dim3(NPTS / 64, DIM / 16, BATCH), wblk, 0, stream>>>(
      w1, seed_feat, bufA, NPTS, (long)DIM * NPTS, (long)272 * NPTS, sc1, sh1);
  gemm_wmma<64, true><<<dim3(NPTS / 64, DIM / 16, BATCH), wblk, 0, stream>>>(
      w2, bufA, bufB, NPTS, (long)272 * NPTS, (long)DIM * NPTS, sc2, sh2);
  gemm_wmma<64, false><<<dim3(NPTS / 64, 272 / 16, BATCH), wblk, 0, stream>>>(
      w3p, bufB, bufA, NPTS, (long)DIM * NPTS, (long)272 * NPTS, sc3, sh3);

  vote_epilogue<<<dim3(NPTS, BATCH), dim3(256), 0, stream>>>(
      bufA, seed_xyz, seed_feat, vxyz, vf);

  fps_kernel<<<dim3(BATCH), dim3(256), 0, stream>>>(vxyz, out_newxyz);

  ball_query_kernel<<<dim3(NPOINT, BATCH), dim3(32), 0, stream>>>(
      vxyz, out_newxyz, bidx);

  gather_group_kernel<<<dim3(NPOINT, BATCH), dim3(256), 0, stream>>>(
      vxyz, vf, out_newxyz, bidx, gbuf);

  // --- stage 2: MLP over 4096 grouped positions ---------------------------
  gemm_wmma<65, true><<<dim3(M2 / 64, 128 / 16, BATCH), wblk, 0, stream>>>(
      mw0p, gbuf, h0, M2, (long)260 * M2, (long)128 * M2, scm0, shm0);
  gemm_wmma<32, true><<<dim3(M2 / 64, 128 / 16, BATCH), wblk, 0, stream>>>(
      mw1, h0, h1, M2, (long)128 * M2, (long)128 * M2, scm1, shm1);
  gemm_wmma<32, true><<<dim3(M2 / 64, DIM / 16, BATCH), wblk, 0, stream>>>(
      mw2, h1, h2, M2, (long)128 * M2, (long)DIM * M2, scm2, shm2);

  maxpool_kernel<<<dim3((BATCH * DIM * NPOINT + 255) / 256), dim3(256), 0, stream>>>(
      h2, out_feat);
}